// GlobalAttention_86964497809566
// MI455X (gfx1250) — compile-verified
//
#include <hip/hip_runtime.h>

typedef __attribute__((ext_vector_type(2))) float v2f;
typedef __attribute__((ext_vector_type(8))) float v8f;

#define LS 33  // LDS row stride (floats), padded to avoid bank conflicts

__device__ __forceinline__ v8f wmma4(v2f a, v2f b, v8f c) {
  // D = A(16x4) * B(4x16) + C, exact fp32 on the matrix pipe
  return __builtin_amdgcn_wmma_f32_16x16x4_f32(false, a, false, b, (short)0, c,
                                               false, false);
}

// A-fragment (16x4) from an LDS matrix, row-major stride LS.
// lanes 0-15: row m, K = kBase+{0,1}; lanes 16-31: row m, K = kBase+{2,3}
__device__ __forceinline__ v2f afrag(const float* M, int mTile, int kBase, int lane) {
  int m  = mTile * 16 + (lane & 15);
  int k0 = kBase + ((lane >> 4) << 1);
  v2f a;
  a.x = M[m * LS + k0];
  a.y = M[m * LS + k0 + 1];
  return a;
}

// B-fragment (4x16) for  X @ W^T : B[k][n] = W[n][k], W row-major [N][32] in global
__device__ __forceinline__ v2f bfragWT(const float* W, int nTile, int kStep, int lane) {
  int n  = nTile * 16 + (lane & 15);
  int k0 = kStep * 4 + ((lane >> 4) << 1);
  v2f b;
  b.x = W[n * 32 + k0];
  b.y = W[n * 32 + k0 + 1];
  return b;
}

// B-fragment for Q_h @ K_h^T : B[d][j] = K[j][h8+d]  (K in LDS)
__device__ __forceinline__ v2f bfragKT(const float* Kb, int nTile, int kStep, int h8, int lane) {
  int j  = nTile * 16 + (lane & 15);
  int k0 = kStep * 4 + ((lane >> 4) << 1);
  v2f b;
  b.x = Kb[j * LS + h8 + k0];
  b.y = Kb[j * LS + h8 + k0 + 1];
  return b;
}

// B-fragment for P @ V_h : B[k][d] = V[k][h8+d], valid d = 0..7, zero-pad d = 8..15
__device__ __forceinline__ v2f bfragV(const float* Vb, int kStep, int h8, int lane) {
  int j  = lane & 15;
  int k0 = kStep * 4 + ((lane >> 4) << 1);
  v2f b;
  if (j < 8) {
    b.x = Vb[k0 * LS + h8 + j];
    b.y = Vb[(k0 + 1) * LS + h8 + j];
  } else {
    b.x = 0.f;
    b.y = 0.f;
  }
  return b;
}

__launch_bounds__(64)
__global__ void ga_kernel(const float* __restrict__ x,
                          const float* __restrict__ qw, const float* __restrict__ qb,
                          const float* __restrict__ kvw, const float* __restrict__ kvb,
                          const float* __restrict__ pw, const float* __restrict__ pb,
                          float* __restrict__ out) {
  __shared__ float sXA[2][32 * LS];  // X, later reused as attention-prob buffer
  __shared__ float sQ[2][32 * LS];
  __shared__ float sK[2][32 * LS];
  __shared__ float sV[2][32 * LS];
  __shared__ float sS[2][32 * LS];   // scrambled pre-proj tensor, indexed by c'

  const int b    = blockIdx.x;
  const int c    = threadIdx.x >> 5;   // wave id = channel
  const int lane = threadIdx.x & 31;

  // weights are hot in L2/WGP$ across all 8192 blocks
  __builtin_prefetch(qw, 0, 3);
  __builtin_prefetch(kvw, 0, 3);
  __builtin_prefetch(pw, 0, 3);

  float* X = sXA[c];
  float* Q = sQ[c];
  float* K = sK[c];
  float* V = sV[c];

  const float* xg = x + (size_t)(b * 2 + c) * 1024;

  // ---- load X[32][32] (coalesced float4) ----
#pragma unroll
  for (int i = 0; i < 8; ++i) {
    int f = i * 128 + lane * 4;
    float4 v = *(const float4*)(xg + f);
    int r = f >> 5, cc = f & 31;
    float* p = &X[r * LS + cc];
    p[0] = v.x; p[1] = v.y; p[2] = v.z; p[3] = v.w;
  }
  __syncthreads();

  const int l15   = lane & 15;
  const int half8 = (lane >> 4) << 3;  // D-frag row offset: 0 or 8

  // ---- Q = X @ qw^T + qb ----
  // Two m-tiles fused per n-tile: shared B-fragment, two interleaved wmma chains,
  // only 16 accumulator VGPRs live (keeps total allocation <= 256 VGPRs).
#pragma unroll 1
  for (int nT = 0; nT < 2; ++nT) {
    float bias = qb[nT * 16 + l15];
    v8f a0 = {}, a1 = {};
#pragma unroll
    for (int k = 0; k < 8; ++k) {
      v2f bf = bfragWT(qw, nT, k, lane);
      a0 = wmma4(afrag(X, 0, k * 4, lane), bf, a0);
      a1 = wmma4(afrag(X, 1, k * 4, lane), bf, a1);
    }
    int n = nT * 16 + l15;
#pragma unroll
    for (int g = 0; g < 8; ++g) {
      Q[(half8 + g) * LS + n]      = a0[g] + bias;
      Q[(16 + half8 + g) * LS + n] = a1[g] + bias;
    }
  }

  // ---- KV = X @ kvw^T + kvb ;  K = cols 0..31, V = cols 32..63 ----
#pragma unroll 1
  for (int nT = 0; nT < 4; ++nT) {
    float bias  = kvb[nT * 16 + l15];
    float* dst  = (nT < 2) ? K : V;
    int    nLoc = (nT & 1) * 16 + l15;
    v8f a0 = {}, a1 = {};
#pragma unroll
    for (int k = 0; k < 8; ++k) {
      v2f bf = bfragWT(kvw, nT, k, lane);
      a0 = wmma4(afrag(X, 0, k * 4, lane), bf, a0);
      a1 = wmma4(afrag(X, 1, k * 4, lane), bf, a1);
    }
#pragma unroll
    for (int g = 0; g < 8; ++g) {
      dst[(half8 + g) * LS + nLoc]      = a0[g] + bias;
      dst[(16 + half8 + g) * LS + nLoc] = a1[g] + bias;
    }
  }
  __syncthreads();

  const float SCALE = 0.35355339059327373f;  // 8^-0.5
  float* P = X;  // reuse X buffer for attention probabilities

#pragma unroll 1
  for (int head = 0; head < 4; ++head) {
    int h8 = head * 8;

    // ---- S = Q_h @ K_h^T  (K-dim = 8 -> 2 wmma per tile) ----
#pragma unroll 1
    for (int nT = 0; nT < 2; ++nT) {
      v8f a0 = {}, a1 = {};
#pragma unroll
      for (int k = 0; k < 2; ++k) {
        v2f bf = bfragKT(K, nT, k, h8, lane);
        a0 = wmma4(afrag(Q, 0, h8 + k * 4, lane), bf, a0);
        a1 = wmma4(afrag(Q, 1, h8 + k * 4, lane), bf, a1);
      }
      int n = nT * 16 + l15;
#pragma unroll
      for (int g = 0; g < 8; ++g) {
        P[(half8 + g) * LS + n]      = a0[g];
        P[(16 + half8 + g) * LS + n] = a1[g];
      }
    }
    __syncthreads();

    // ---- softmax: one row per lane (32 rows, 32 lanes) ----
    {
      float* row = &P[lane * LS];
      float mx = -3.4e38f;
#pragma unroll
      for (int j = 0; j < 32; ++j) mx = fmaxf(mx, row[j] * SCALE);
      float s = 0.f;
#pragma unroll
      for (int j = 0; j < 32; ++j) {
        float e = __expf(row[j] * SCALE - mx);
        row[j] = e;
        s += e;
      }
      float inv = 1.f / s;
#pragma unroll
      for (int j = 0; j < 32; ++j) row[j] *= inv;
    }
    __syncthreads();

    // ---- O_h = P @ V_h ; scatter into scrambled buffer ----
    // scramble: (n,c,h,d) -> c'=n>>1, h'=(n&1)*16 + c*8 + (h>>2), w'=(h&3)*8 + d
    {
      v8f a0 = {}, a1 = {};
#pragma unroll
      for (int k = 0; k < 8; ++k) {
        v2f bf = bfragV(V, k, h8, lane);
        a0 = wmma4(afrag(P, 0, k * 4, lane), bf, a0);
        a1 = wmma4(afrag(P, 1, k * 4, lane), bf, a1);
      }
      if (l15 < 8) {
        int d  = l15;
        int cp = head >> 1;
        int hb = (head & 1) * 16 + c * 8;
#pragma unroll
        for (int g = 0; g < 8; ++g) {
          int h0 = half8 + g;        // m-tile 0 row
          int h1 = 16 + half8 + g;   // m-tile 1 row
          sS[cp][(hb + (h0 >> 2)) * LS + (h0 & 3) * 8 + d] = a0[g];
          sS[cp][(hb + (h1 >> 2)) * LS + (h1 & 3) * 8 + d] = a1[g];
        }
      }
    }
    __syncthreads();
  }

  // ---- proj: Y = S_{c'} @ pw^T + pb ; wave c handles c' = c ----
  const float* S = sS[c];
  float* og = out + (size_t)(b * 2 + c) * 1024;
#pragma unroll 1
  for (int nT = 0; nT < 2; ++nT) {
    float bias = pb[nT * 16 + l15];
    v8f a0 = {}, a1 = {};
#pragma unroll
    for (int k = 0; k < 8; ++k) {
      v2f bf = bfragWT(pw, nT, k, lane);
      a0 = wmma4(afrag(S, 0, k * 4, lane), bf, a0);
      a1 = wmma4(afrag(S, 1, k * 4, lane), bf, a1);
    }
    int n = nT * 16 + l15;
#pragma unroll
    for (int g = 0; g < 8; ++g) {
      og[(half8 + g) * 32 + n]      = a0[g] + bias;
      og[(16 + half8 + g) * 32 + n] = a1[g] + bias;
    }
  }
}

extern "C" void kernel_launch(void* const* d_in, const int* in_sizes, int n_in,
                              void* d_out, int out_size, void* d_ws, size_t ws_size,
                              hipStream_t stream) {
  const float* x   = (const float*)d_in[0];
  const float* qw  = (const float*)d_in[1];
  const float* qb  = (const float*)d_in[2];
  const float* kvw = (const float*)d_in[3];
  const float* kvb = (const float*)d_in[4];
  const float* pw  = (const float*)d_in[5];
  const float* pb  = (const float*)d_in[6];
  float* out = (float*)d_out;

  int B = in_sizes[0] / (2 * 32 * 32);  // 8192
  ga_kernel<<<B, 64, 0, stream>>>(x, qw, qb, kvw, kvb, pw, pb, out);

  (void)n_in; (void)out_size; (void)d_ws; (void)ws_size;
}